// DocREModel_48430051229943
// MI455X (gfx1250) — compile-verified
//
#include <hip/hip_runtime.h>
#include <math.h>

#define N_ 256
#define D_ 768
#define E_ 768
#define F_ 96
#define H_ 8
#define L_ 2
#define ALPHA_ 0.2f
#define NEGV_ -9000000000000000.0f
#define NN_ 65536               // N_*N_
#define OUTLD_ ((L_ + 1) * D_)  // 2304

typedef __attribute__((ext_vector_type(2))) float v2f;
typedef __attribute__((ext_vector_type(8))) float v8f;

// D = A(16x4 f32) * B(4x16 f32) + C(16x16 f32)
__device__ __forceinline__ v8f wmma4(v2f a, v2f b, v8f c) {
  return __builtin_amdgcn_wmma_f32_16x16x4_f32(false, a, false, b, (short)0, c,
                                               false, false);
}

// ---- copy x into first D_ columns of output ----------------------------------
__global__ void k_copyx(const float* __restrict__ x, float* __restrict__ dout) {
  int t = blockIdx.x * 256 + threadIdx.x;  // N_*D_ threads
  int n = t / D_, d = t % D_;
  dout[(long)n * OUTLD_ + d] = x[t];
}

// ---- u[l,h,d] = sum_f Wk[l,h,D_+d,f] * a2[l,h,f]; stored row-major (E_ x 16) --
__global__ void k_prep_u(const float* __restrict__ Wk, const float* __restrict__ a,
                         float* __restrict__ U) {
  int t = blockIdx.x * 256 + threadIdx.x;  // L_*H_*E_ threads
  if (t >= L_ * H_ * E_) return;
  int d = t % E_;
  int lh = t / E_;
  const float* w = Wk + (size_t)lh * (D_ + E_) * F_ + (size_t)(D_ + d) * F_;
  const float* a2 = a + (size_t)lh * 2 * F_ + F_;
  float s = 0.f;
  for (int f = 0; f < F_; ++f) s += w[f] * a2[f];
  U[(size_t)d * (L_ * H_) + lh] = s;
}

// ---- edge scores: (NN_ x E_) @ (E_ x 16) for both layers in one pass over e --
// LDS-staged: coalesced float4 global loads -> LDS tile -> ds_load WMMA frags.
#define KC_ 32    // k-chunk (columns per LDS tile)
#define ROWS_ 128 // e-rows per block (8 waves x 16)
#define PAD_ 36   // LDS row stride in floats: 16B-aligned rows, bank-conflict-free
__global__ void __launch_bounds__(256) k_edge_scores(const float* __restrict__ e,
                                                     const float* __restrict__ U,
                                                     float* __restrict__ s_e) {
  __shared__ float Ulds[E_ * 16];        // 48 KB, B-matrix (K x 16)
  __shared__ float Atile[ROWS_ * PAD_];  // 18 KB, staged e tile
  int tid = threadIdx.x;
  for (int t = tid; t < E_ * 16; t += 256) Ulds[t] = U[t];
  int wave = tid >> 5, lane = tid & 31;
  int lid = lane & 15, half = lane >> 4;
  long rbase = (long)blockIdx.x * ROWS_;
  const float* Eb = e + rbase * E_;
  v8f c = {};
  for (int k0 = 0; k0 < E_; k0 += KC_) {
    __syncthreads();  // Atile safe to overwrite; also covers Ulds readiness
#pragma unroll
    for (int p = 0; p < 4; ++p) {
      int idx = tid + p * 256;  // 0..1023 quads
      int row = idx >> 3;       // 128 rows, 8 quads each
      int c4 = (idx & 7) * 4;   // coalesced 128B per row across 8 lanes
      float4 v = *(const float4*)(Eb + (long)row * E_ + k0 + c4);
      *(float4*)(Atile + row * PAD_ + c4) = v;
    }
    __syncthreads();
    const float* Arow = Atile + (wave * 16 + lid) * PAD_;
#pragma unroll
    for (int kk = 0; kk < KC_; kk += 4) {
      int k2 = kk + half * 2;
      v2f av, bv;
      av.x = Arow[k2];
      av.y = Arow[k2 + 1];
      bv.x = Ulds[(k0 + k2) * 16 + lid];
      bv.y = Ulds[(k0 + k2 + 1) * 16 + lid];
      c = wmma4(av, bv, c);
    }
  }
  long mrow = rbase + wave * 16;
#pragma unroll
  for (int r = 0; r < 8; ++r) {
    long m = mrow + r + half * 8;
    s_e[(long)lid * NN_ + m] = c[r];  // lid = l*H_+h
  }
}

// ---- node projections: hq = cur@Wq[l,h], hkx = cur@Wk_x[l,h], hvx = cur@Wv_x[l,h]
__global__ void k_proj(const float* __restrict__ cur, long clda,
                       const float* __restrict__ Wq, const float* __restrict__ Wk,
                       const float* __restrict__ Wv, int layer,
                       float* __restrict__ hq, float* __restrict__ hkx,
                       float* __restrict__ hvx) {
  int wave = threadIdx.x >> 5;
  int lane = threadIdx.x & 31;
  int lid = lane & 15, half = lane >> 4;
  int W = blockIdx.x * 8 + wave;  // 3*8*6*16 = 2304 waves
  int mt = W & 15; W >>= 4;
  int nt = W % 6; W /= 6;
  int h = W & 7;
  int mat = W >> 3;
  const float* A = cur + (long)(mt * 16) * clda;
  const float* B;
  float* O;
  if (mat == 0) {
    B = Wq + (size_t)(layer * H_ + h) * D_ * F_;
    O = hq;
  } else if (mat == 1) {
    B = Wk + (size_t)(layer * H_ + h) * (D_ + E_) * F_;
    O = hkx;
  } else {
    B = Wv + (size_t)(layer * H_ + h) * (D_ + E_) * F_;
    O = hvx;
  }
  int nb = nt * 16;
  v8f c = {};
  for (int k0 = 0; k0 < D_; k0 += 4) {
    int kk = k0 + half * 2;
    v2f av, bv;
    av.x = A[(long)lid * clda + kk];
    av.y = A[(long)lid * clda + kk + 1];
    bv.x = B[(long)kk * F_ + nb + lid];
    bv.y = B[(long)(kk + 1) * F_ + nb + lid];
    c = wmma4(av, bv, c);
  }
#pragma unroll
  for (int r = 0; r < 8; ++r) {
    int m = mt * 16 + r + half * 8;
    O[(long)h * N_ * F_ + (long)m * F_ + nb + lid] = c[r];
  }
}

// ---- per-(h,n) dots with attention vectors a1, a2 ----------------------------
__global__ void k_dots(const float* __restrict__ hq, const float* __restrict__ hkx,
                       const float* __restrict__ a, int layer,
                       float* __restrict__ qd, float* __restrict__ kd) {
  int t = blockIdx.x * 256 + threadIdx.x;  // H_*N_ threads
  if (t >= H_ * N_) return;
  int h = t / N_;
  const float* a1 = a + (size_t)(layer * H_ + h) * 2 * F_;
  const float* a2 = a1 + F_;
  const float* q = hq + (long)t * F_;
  const float* k = hkx + (long)t * F_;
  float s1 = 0.f, s2 = 0.f;
  for (int f = 0; f < F_; ++f) {
    s1 += q[f] * a1[f];
    s2 += k[f] * a2[f];
  }
  qd[t] = s1;
  kd[t] = s2;
}

// ---- masked leaky-ReLU softmax over j; one block per (h,i) -------------------
__global__ void k_softmax(const float* __restrict__ qd, const float* __restrict__ kd,
                          const float* __restrict__ s_e, const int* __restrict__ adj,
                          int layer, float* __restrict__ att) {
  __shared__ float red[256];
  int h = blockIdx.x / N_;
  int i = blockIdx.x % N_;
  int j = threadIdx.x;
  float s = qd[h * N_ + i] + kd[h * N_ + j] +
            s_e[(long)(layer * H_ + h) * NN_ + (long)i * N_ + j];
  s = s > 0.f ? s : ALPHA_ * s;
  s = (adj[(long)i * N_ + j] > 0) ? s : NEGV_;
  red[j] = s;
  __syncthreads();
  for (int off = 128; off > 0; off >>= 1) {
    if (j < off) red[j] = fmaxf(red[j], red[j + off]);
    __syncthreads();
  }
  float mx = red[0];
  __syncthreads();
  float ex = __expf(s - mx);
  red[j] = ex;
  __syncthreads();
  for (int off = 128; off > 0; off >>= 1) {
    if (j < off) red[j] += red[j + off];
    __syncthreads();
  }
  att[(long)h * NN_ + (long)i * N_ + j] = ex / red[0];
}

// ---- we[i,h,:] = sum_j att[h,i,j] * e[i,j,:]  (batched 16x16 WMMA, 8 real rows)
__global__ void k_wesum(const float* __restrict__ att, const float* __restrict__ e,
                        float* __restrict__ we) {
  int wave = threadIdx.x >> 5;
  int lane = threadIdx.x & 31;
  int lid = lane & 15, half = lane >> 4;
  int i = blockIdx.x / 6;
  int nb = ((blockIdx.x % 6) * 8 + wave) * 16;
  const float* B = e + (long)i * N_ * E_ + nb;
  bool arow = lid < H_;
  const float* Arow = att + (long)(arow ? lid : 0) * NN_ + (long)i * N_;
  v8f c = {};
  for (int k0 = 0; k0 < N_; k0 += 4) {
    int kk = k0 + half * 2;
    v2f av, bv;
    av.x = arow ? Arow[kk] : 0.f;
    av.y = arow ? Arow[kk + 1] : 0.f;
    bv.x = B[(long)kk * E_ + lid];
    bv.y = B[(long)(kk + 1) * E_ + lid];
    c = wmma4(av, bv, c);
  }
#pragma unroll
  for (int r = 0; r < 8; ++r) {
    int m = r + half * 8;
    if (m < H_) we[((long)i * H_ + m) * E_ + nb + lid] = c[r];
  }
}

// ---- out = elu(att@hvx + we@Wv_e); write into d_out columns of this layer ----
__global__ void k_out(const float* __restrict__ att, const float* __restrict__ hvx,
                      const float* __restrict__ we, const float* __restrict__ Wv,
                      int layer, float* __restrict__ dout) {
  int wave = threadIdx.x >> 5;
  int lane = threadIdx.x & 31;
  int lid = lane & 15, half = lane >> 4;
  int W = blockIdx.x * 8 + wave;  // 8*6*16 = 768 waves
  int mt = W & 15; W >>= 4;
  int nt = W % 6;
  int h = W / 6;
  int mb = mt * 16, nb = nt * 16;
  v8f c = {};
  // phase 1: att[h] (256x256) @ hvx[h] (256x96)
  const float* A1 = att + (long)h * NN_ + (long)mb * N_;
  const float* B1 = hvx + (long)h * N_ * F_ + nb;
  for (int k0 = 0; k0 < N_; k0 += 4) {
    int kk = k0 + half * 2;
    v2f av, bv;
    av.x = A1[(long)lid * N_ + kk];
    av.y = A1[(long)lid * N_ + kk + 1];
    bv.x = B1[(long)kk * F_ + lid];
    bv.y = B1[(long)(kk + 1) * F_ + lid];
    c = wmma4(av, bv, c);
  }
  // phase 2: we[:,h,:] (256x768) @ Wv_e[l,h] (768x96)
  const float* A2 = we + (long)h * E_;  // row stride H_*E_
  const float* B2 = Wv + (size_t)(layer * H_ + h) * (D_ + E_) * F_ + (size_t)D_ * F_ + nb;
  for (int k0 = 0; k0 < E_; k0 += 4) {
    int kk = k0 + half * 2;
    v2f av, bv;
    av.x = A2[(long)(mb + lid) * (H_ * E_) + kk];
    av.y = A2[(long)(mb + lid) * (H_ * E_) + kk + 1];
    bv.x = B2[(long)kk * F_ + lid];
    bv.y = B2[(long)(kk + 1) * F_ + lid];
    c = wmma4(av, bv, c);
  }
#pragma unroll
  for (int r = 0; r < 8; ++r) {
    int m = mb + r + half * 8;
    float v = c[r];
    v = v > 0.f ? v : (__expf(v) - 1.f);  // ELU (alpha=1)
    dout[(long)m * OUTLD_ + (long)(layer + 1) * D_ + h * F_ + nb + lid] = v;
  }
}

extern "C" void kernel_launch(void* const* d_in, const int* in_sizes, int n_in,
                              void* d_out, int out_size, void* d_ws, size_t ws_size,
                              hipStream_t stream) {
  (void)in_sizes; (void)n_in; (void)out_size; (void)ws_size;
  const float* x = (const float*)d_in[0];
  const int* adj = (const int*)d_in[1];
  const float* e = (const float*)d_in[2];
  const float* Wq = (const float*)d_in[3];
  const float* Wk = (const float*)d_in[4];
  const float* Wv = (const float*)d_in[5];
  const float* a = (const float*)d_in[6];
  float* dout = (float*)d_out;

  // workspace partition (floats); ~15 MB total
  float* ws = (float*)d_ws;
  float* U = ws;                          // E_ * L_*H_          = 12288
  float* s_e = U + (size_t)E_ * L_ * H_;  // L_*H_*NN_           = 1048576
  float* hq = s_e + (size_t)L_ * H_ * NN_;  // H_*N_*F_          = 196608
  float* hkx = hq + (size_t)H_ * N_ * F_;
  float* hvx = hkx + (size_t)H_ * N_ * F_;
  float* qd = hvx + (size_t)H_ * N_ * F_;  // H_*N_              = 2048
  float* kd = qd + (size_t)H_ * N_;
  float* att = kd + (size_t)H_ * N_;  // H_*NN_                  = 524288
  float* we = att + (size_t)H_ * NN_; // N_*H_*E_                = 1572864

  k_copyx<<<(N_ * D_) / 256, 256, 0, stream>>>(x, dout);
  k_prep_u<<<(L_ * H_ * E_ + 255) / 256, 256, 0, stream>>>(Wk, a, U);
  k_edge_scores<<<NN_ / ROWS_, 256, 0, stream>>>(e, U, s_e);

  for (int l = 0; l < L_; ++l) {
    const float* cur = (l == 0) ? x : (dout + D_);
    long clda = (l == 0) ? D_ : OUTLD_;
    k_proj<<<(3 * H_ * 6 * 16) / 8, 256, 0, stream>>>(cur, clda, Wq, Wk, Wv, l,
                                                      hq, hkx, hvx);
    k_dots<<<(H_ * N_ + 255) / 256, 256, 0, stream>>>(hq, hkx, a, l, qd, kd);
    k_softmax<<<H_ * N_, 256, 0, stream>>>(qd, kd, s_e, adj, l, att);
    k_wesum<<<N_ * 6, 256, 0, stream>>>(att, e, we);
    k_out<<<(H_ * 6 * 16) / 8, 256, 0, stream>>>(att, hvx, we, Wv, l, dout);
  }
}